// MSCAA_27831388078269
// MI455X (gfx1250) — compile-verified
//
#include <hip/hip_runtime.h>

// MSCA (SegNeXt multi-scale conv attention), fully fused per (batch,channel) plane.
// Layout: one workgroup per 64x64 plane; whole conv chain lives in LDS.
// CDNA5-specific path: global_load_async_to_lds_b128 (ASYNCcnt) for the input stage.

#define CCH        256
#define BATCH      8
#define HH         64
#define WW         64
#define PW         12            // horizontal guard (>=10, mult of 4 for b128 alignment)
#define PH         10            // vertical guard (>=10)
#define LSTRIDE    (WW + 2*PW)   // 88 floats per LDS row (mult of 4 -> rows stay 16B aligned)
#define LROWS      (HH + 2*PH)   // 84
#define BUF_FLOATS (LROWS * LSTRIDE)       // 7392 floats = 28.875 KB
#define NPLANE     (BATCH * CCH)           // 2048
#define PLANE_PIX  (HH * WW)               // 4096
#define NPIX       ((size_t)NPLANE * PLANE_PIX)

// r in [-PH, HH+PH), c in [-PW, WW+PW)
#define IDX(r, c) (((PH + (r)) * LSTRIDE) + PW + (c))

template <int TAPS>
__device__ __forceinline__ void rowconv16(const float* src, const float* wv, float bias,
                                          int r0, int c, float out[16]) {
#pragma unroll
  for (int k = 0; k < 16; ++k) out[k] = bias;
#pragma unroll
  for (int d = 0; d < TAPS; ++d) {
    const float w = wv[d];
#pragma unroll
    for (int k = 0; k < 16; ++k)
      out[k] = fmaf(w, src[IDX(r0 + k, c + d - TAPS / 2)], out[k]);
  }
}

template <int TAPS>
__device__ __forceinline__ void colconv16(const float* src, const float* wv, float bias,
                                          int r0, int c, float out[16]) {
#pragma unroll
  for (int k = 0; k < 16; ++k) out[k] = bias;
#pragma unroll
  for (int d = 0; d < TAPS; ++d) {
    const float w = wv[d];
#pragma unroll
    for (int k = 0; k < 16; ++k)
      out[k] = fmaf(w, src[IDX(r0 + k + d - TAPS / 2, c)], out[k]);
  }
}

__global__ __launch_bounds__(256)
void msca_fused(const float* __restrict__ x,
                const float* __restrict__ w0,   const float* __restrict__ b0,
                const float* __restrict__ w0_1, const float* __restrict__ b0_1,
                const float* __restrict__ w0_2, const float* __restrict__ b0_2,
                const float* __restrict__ w1_1, const float* __restrict__ b1_1,
                const float* __restrict__ w1_2, const float* __restrict__ b1_2,
                const float* __restrict__ w2_1, const float* __restrict__ b2_1,
                const float* __restrict__ w2_2, const float* __restrict__ b2_2,
                float* __restrict__ out0, float* __restrict__ out1, float* __restrict__ out2)
{
  __shared__ __align__(16) float lds[4 * BUF_FLOATS];
  float* B0 = lds;                       // X            -> later A1X
  float* B1 = lds + 1 * BUF_FLOATS;      // ATTN (5x5)   -> later A1Y
  float* B2 = lds + 2 * BUF_FLOATS;      // A0X
  float* B3 = lds + 3 * BUF_FLOATS;      // A0Y

  const int t     = threadIdx.x;
  const int plane = blockIdx.x;          // b*256 + ch
  const int ch    = plane & (CCH - 1);
  const int c     = t & 63;              // lane-consecutive columns: conflict-free DS
  const int r0    = (t >> 6) << 4;       // 16-row vertical strip per thread

  // ---- zero ALL of LDS once: guard cells implement the conv zero-padding ----
  {
    const float4 z = make_float4(0.f, 0.f, 0.f, 0.f);
    float4* l4 = reinterpret_cast<float4*>(lds);
#pragma unroll 4
    for (int i = t; i < (4 * BUF_FLOATS) / 4; i += 256) l4[i] = z;
  }
  __syncthreads();

  // ---- CDNA5 async copy: 64x64 input plane -> B0 interior (16B-aligned b128 per lane) ----
  {
    const float* xp = x + (size_t)plane * PLANE_PIX;
#pragma unroll
    for (int it = 0; it < 4; ++it) {
      const int chunk = t + it * 256;              // 0..1023, 16B each
      const int row   = chunk >> 4;                // 0..63
      const int cc    = (chunk & 15) << 2;         // 0,4,...,60
      const unsigned  ldsoff = (unsigned)(IDX(row, cc) * 4);  // byte offset, 16B aligned
      const float*    gp     = xp + row * WW + cc;
      asm volatile("global_load_async_to_lds_b128 %0, %1, off"
                   :: "v"(ldsoff), "v"(gp) : "memory");
    }
    asm volatile("s_wait_asynccnt 0" ::: "memory");
  }
  __syncthreads();

  // ---- Stage A: 5x5 depthwise (pad 2), B0 -> B1 ----
  {
    float wA[25];
#pragma unroll
    for (int i = 0; i < 25; ++i) wA[i] = w0[ch * 25 + i];
    const float bA = b0[ch];
#pragma unroll
    for (int half = 0; half < 2; ++half) {
      float acc[8];
#pragma unroll
      for (int k = 0; k < 8; ++k) acc[k] = bA;
      const int rb = r0 + half * 8;
#pragma unroll
      for (int dy = 0; dy < 5; ++dy)
#pragma unroll
        for (int dx = 0; dx < 5; ++dx) {
          const float w = wA[dy * 5 + dx];
#pragma unroll
          for (int k = 0; k < 8; ++k)
            acc[k] = fmaf(w, B0[IDX(rb + k + dy - 2, c + dx - 2)], acc[k]);
        }
#pragma unroll
      for (int k = 0; k < 8; ++k) B1[IDX(rb + k, c)] = acc[k];
    }
  }
  __syncthreads();

  // ---- Stage 0: 1x7 / 7x1 on ATTN; attn_0 = a0x*a0y ----
  {
    float wx[7], wy[7];
#pragma unroll
    for (int i = 0; i < 7; ++i) { wx[i] = w0_1[ch * 7 + i]; wy[i] = w0_2[ch * 7 + i]; }
    float ax[16], ay[16];
    rowconv16<7>(B1, wx, b0_1[ch], r0, c, ax);
    colconv16<7>(B1, wy, b0_2[ch], r0, c, ay);
    float* o0 = out0 + (size_t)plane * PLANE_PIX;
#pragma unroll
    for (int k = 0; k < 16; ++k) {
      B2[IDX(r0 + k, c)] = ax[k];
      B3[IDX(r0 + k, c)] = ay[k];
      o0[(r0 + k) * WW + c] = ax[k] * ay[k];
    }
  }
  __syncthreads();

  // ---- Stage 1: 1x11 on A0X -> B0, 11x1 on A0Y -> B1; attn_1 = a1x*a1y ----
  {
    float wx[11], wy[11];
#pragma unroll
    for (int i = 0; i < 11; ++i) { wx[i] = w1_1[ch * 11 + i]; wy[i] = w1_2[ch * 11 + i]; }
    float ax[16], ay[16];
    rowconv16<11>(B2, wx, b1_1[ch], r0, c, ax);
    colconv16<11>(B3, wy, b1_2[ch], r0, c, ay);
    float* o1 = out1 + (size_t)plane * PLANE_PIX;
#pragma unroll
    for (int k = 0; k < 16; ++k) {
      B0[IDX(r0 + k, c)] = ax[k];     // interiors rewritten; guards stay zero
      B1[IDX(r0 + k, c)] = ay[k];
      o1[(r0 + k) * WW + c] = ax[k] * ay[k];
    }
  }
  __syncthreads();

  // ---- Stage 2: 1x21 on A1X, 21x1 on A1Y; attn_2 = a2x*a2y (no LDS stores) ----
  {
    float wx[21], wy[21];
#pragma unroll
    for (int i = 0; i < 21; ++i) { wx[i] = w2_1[ch * 21 + i]; wy[i] = w2_2[ch * 21 + i]; }
    float ax[16], ay[16];
    rowconv16<21>(B0, wx, b2_1[ch], r0, c, ax);
    colconv16<21>(B1, wy, b2_2[ch], r0, c, ay);
    float* o2 = out2 + (size_t)plane * PLANE_PIX;
#pragma unroll
    for (int k = 0; k < 16; ++k)
      o2[(r0 + k) * WW + c] = ax[k] * ay[k];
  }
}

extern "C" void kernel_launch(void* const* d_in, const int* in_sizes, int n_in,
                              void* d_out, int out_size, void* d_ws, size_t ws_size,
                              hipStream_t stream) {
  const float* x    = (const float*)d_in[0];
  const float* w0   = (const float*)d_in[1];
  const float* b0   = (const float*)d_in[2];
  const float* w0_1 = (const float*)d_in[3];
  const float* b0_1 = (const float*)d_in[4];
  const float* w0_2 = (const float*)d_in[5];
  const float* b0_2 = (const float*)d_in[6];
  const float* w1_1 = (const float*)d_in[7];
  const float* b1_1 = (const float*)d_in[8];
  const float* w1_2 = (const float*)d_in[9];
  const float* b1_2 = (const float*)d_in[10];
  const float* w2_1 = (const float*)d_in[11];
  const float* b2_1 = (const float*)d_in[12];
  const float* w2_2 = (const float*)d_in[13];
  const float* b2_2 = (const float*)d_in[14];
  float* out = (float*)d_out;

  msca_fused<<<NPLANE, 256, 0, stream>>>(
      x, w0, b0, w0_1, b0_1, w0_2, b0_2,
      w1_1, b1_1, w1_2, b1_2, w2_1, b2_1, w2_2, b2_2,
      out, out + NPIX, out + 2 * NPIX);

  (void)in_sizes; (void)n_in; (void)out_size; (void)d_ws; (void)ws_size;
}